// GCN_66511863546049
// MI455X (gfx1250) — compile-verified
//
#include <hip/hip_runtime.h>
#include <math.h>

typedef __attribute__((ext_vector_type(2))) float v2f;
typedef __attribute__((ext_vector_type(8))) float v8f;

#define TPB 256
#define FIN 512
#define HID 16
#define NCLS 7

// ---------------- init: deg=1 (self loop), zero accumulators ----------------
__global__ void gcn_init_kernel(float* __restrict__ deg, float* __restrict__ acc1,
                                float* __restrict__ acc2, int n) {
    int idx = blockIdx.x * blockDim.x + threadIdx.x;
    if (idx < n) deg[idx] = 1.0f;
    if (idx < 16 * n) acc1[idx] = 0.0f;
    if (idx < 8 * n) acc2[idx] = 0.0f;
}

// ---------------- degree: deg[dst] += 1 per edge ----------------
__global__ void gcn_degree_kernel(const int* __restrict__ dst, float* __restrict__ deg,
                                  long long E) {
    long long e = (long long)blockIdx.x * blockDim.x + threadIdx.x;
    if (e >= E) return;
    atomicAdd(deg + dst[e], 1.0f);
}

// ---------------- dinv = rsqrt(max(deg,1)) in place ----------------
__global__ void gcn_rsqrt_kernel(float* __restrict__ deg, int n) {
    int i = blockIdx.x * blockDim.x + threadIdx.x;
    if (i >= n) return;
    deg[i] = rsqrtf(fmaxf(deg[i], 1.0f));
}

// ---------------- layer-1 GEMM: hlin = x @ W1 via V_WMMA_F32_16X16X4_F32 ----
__global__ __launch_bounds__(TPB) void gcn_gemm1_kernel(const float* __restrict__ x,
                                                        const float* __restrict__ W1,
                                                        float* __restrict__ hlin, int n) {
    __shared__ float Wl[FIN * HID];  // 32 KB
    for (int i = threadIdx.x; i < FIN * HID; i += TPB) Wl[i] = W1[i];
    __syncthreads();

    const int lane = threadIdx.x & 31;
    const int wave = threadIdx.x >> 5;
    const long long tile = (long long)blockIdx.x * (TPB / 32) + wave;
    const long long rowBase = tile * 16;
    if (rowBase >= n) return;

    if (rowBase + 16 <= n) {
        const int r = lane & 15;            // A: row / B: col / D: col
        const int kb = (lane >> 4) << 1;    // lanes 0-15 -> K 0,1 ; lanes 16-31 -> K 2,3
        v8f c = {};
        const float* xp = x + (rowBase + r) * FIN + kb;
        #pragma unroll 4
        for (int ko = 0; ko < FIN / 4; ++ko) {
            v2f a, b;
            float2 av = *(const float2*)(xp + ko * 4);
            a[0] = av.x; a[1] = av.y;
            const int k0 = ko * 4 + kb;
            b[0] = Wl[k0 * HID + r];
            b[1] = Wl[(k0 + 1) * HID + r];
            c = __builtin_amdgcn_wmma_f32_16x16x4_f32(false, a, false, b,
                                                      (short)0, c, false, false);
        }
        const int mB = (lane >> 4) << 3;    // D rows: lanes 0-15 -> M 0..7, 16-31 -> M 8..15
        #pragma unroll
        for (int j = 0; j < 8; ++j)
            hlin[(rowBase + mB + j) * HID + r] = c[j];
    } else {
        long long row = rowBase + lane;     // scalar tail (unused for N%16==0)
        if (row < n) {
            for (int col = 0; col < HID; ++col) {
                float s = 0.0f;
                for (int k = 0; k < FIN; ++k) s += x[row * FIN + k] * Wl[k * HID + col];
                hlin[row * HID + col] = s;
            }
        }
    }
}

// ------ scatter-add layer 1: one thread per edge, b128 gathers, 16 atomics ---
__global__ void gcn_scatter1_kernel(const int* __restrict__ src, const int* __restrict__ dst,
                                    const float* __restrict__ dinv,
                                    const float* __restrict__ hlin,
                                    float* __restrict__ acc1, long long E) {
    long long e = (long long)blockIdx.x * blockDim.x + threadIdx.x;
    if (e >= E) return;
    int s = src[e], d = dst[e];
    float nrm = dinv[s] * dinv[d];
    const float4* hp = (const float4*)(hlin + (long long)s * HID);
    float* ap = acc1 + (long long)d * HID;
    #pragma unroll
    for (int q = 0; q < 4; ++q) {
        float4 v = hp[q];
        atomicAdd(ap + q * 4 + 0, v.x * nrm);
        atomicAdd(ap + q * 4 + 1, v.y * nrm);
        atomicAdd(ap + q * 4 + 2, v.z * nrm);
        atomicAdd(ap + q * 4 + 3, v.w * nrm);
    }
}

// ---------------- self-loop + bias + relu (in place on acc1) ----------------
__global__ void gcn_post1_kernel(const float* __restrict__ dinv,
                                 const float* __restrict__ hlin,
                                 const float* __restrict__ b1,
                                 float* __restrict__ acc1, int n) {
    long long t = (long long)blockIdx.x * blockDim.x + threadIdx.x;
    long long i = t >> 4;
    int f = (int)(t & 15);
    if (i >= n) return;
    float di = dinv[i];
    float v = acc1[i * HID + f] + hlin[i * HID + f] * di * di + b1[f];
    acc1[i * HID + f] = fmaxf(v, 0.0f);
}

// ---------------- layer-2 GEMM: h2lin = h1 @ W2pad via WMMA (K=16) -----------
__global__ __launch_bounds__(TPB) void gcn_gemm2_kernel(const float* __restrict__ h1,
                                                        const float* __restrict__ W2,
                                                        float* __restrict__ h2lin, int n) {
    __shared__ float Wl[16 * 16];  // W2 zero-padded from 7 to 16 columns
    {
        int k = threadIdx.x >> 4, nn = threadIdx.x & 15;
        if (threadIdx.x < 256) Wl[threadIdx.x] = (nn < NCLS) ? W2[k * NCLS + nn] : 0.0f;
    }
    __syncthreads();

    const int lane = threadIdx.x & 31;
    const int wave = threadIdx.x >> 5;
    const long long tile = (long long)blockIdx.x * (TPB / 32) + wave;
    const long long rowBase = tile * 16;
    if (rowBase >= n) return;

    if (rowBase + 16 <= n) {
        const int r = lane & 15;
        const int kb = (lane >> 4) << 1;
        v8f c = {};
        const float* hp = h1 + (rowBase + r) * HID + kb;
        #pragma unroll
        for (int ko = 0; ko < HID / 4; ++ko) {
            v2f a, b;
            float2 av = *(const float2*)(hp + ko * 4);
            a[0] = av.x; a[1] = av.y;
            const int k0 = ko * 4 + kb;
            b[0] = Wl[k0 * 16 + r];
            b[1] = Wl[(k0 + 1) * 16 + r];
            c = __builtin_amdgcn_wmma_f32_16x16x4_f32(false, a, false, b,
                                                      (short)0, c, false, false);
        }
        const int mB = (lane >> 4) << 3;
        #pragma unroll
        for (int j = 0; j < 8; ++j)
            h2lin[(rowBase + mB + j) * 16 + r] = c[j];
    } else {
        long long row = rowBase + lane;
        if (row < n) {
            for (int col = 0; col < 16; ++col) {
                float s = 0.0f;
                if (col < NCLS)
                    for (int k = 0; k < HID; ++k) s += h1[row * HID + k] * Wl[k * 16 + col];
                h2lin[row * 16 + col] = s;
            }
        }
    }
}

// ------ scatter-add layer 2: one thread per edge, 2x b128 gathers, 7 atomics -
__global__ void gcn_scatter2_kernel(const int* __restrict__ src, const int* __restrict__ dst,
                                    const float* __restrict__ dinv,
                                    const float* __restrict__ h2lin,
                                    float* __restrict__ acc2, long long E) {
    long long e = (long long)blockIdx.x * blockDim.x + threadIdx.x;
    if (e >= E) return;
    int s = src[e], d = dst[e];
    float nrm = dinv[s] * dinv[d];
    const float4* hp = (const float4*)(h2lin + (long long)s * 16);
    float4 v0 = hp[0];
    float4 v1 = hp[1];
    float* ap = acc2 + (long long)d * 8;
    atomicAdd(ap + 0, v0.x * nrm);
    atomicAdd(ap + 1, v0.y * nrm);
    atomicAdd(ap + 2, v0.z * nrm);
    atomicAdd(ap + 3, v0.w * nrm);
    atomicAdd(ap + 4, v1.x * nrm);
    atomicAdd(ap + 5, v1.y * nrm);
    atomicAdd(ap + 6, v1.z * nrm);
}

// ---------------- self-loop + bias + log_softmax ----------------
__global__ void gcn_final_kernel(const float* __restrict__ dinv,
                                 const float* __restrict__ h2lin,
                                 const float* __restrict__ acc2,
                                 const float* __restrict__ b2,
                                 float* __restrict__ out, int n) {
    long long i = (long long)blockIdx.x * blockDim.x + threadIdx.x;
    if (i >= n) return;
    float di = dinv[i];
    float d2 = di * di;
    const float4* ap = (const float4*)(acc2 + i * 8);
    const float4* hp = (const float4*)(h2lin + i * 16);
    float4 a0 = ap[0], a1 = ap[1];
    float4 h0 = hp[0], h1 = hp[1];
    float v[NCLS];
    v[0] = a0.x + h0.x * d2 + b2[0];
    v[1] = a0.y + h0.y * d2 + b2[1];
    v[2] = a0.z + h0.z * d2 + b2[2];
    v[3] = a0.w + h0.w * d2 + b2[3];
    v[4] = a1.x + h1.x * d2 + b2[4];
    v[5] = a1.y + h1.y * d2 + b2[5];
    v[6] = a1.z + h1.z * d2 + b2[6];
    float m = -INFINITY;
    #pragma unroll
    for (int c = 0; c < NCLS; ++c) m = fmaxf(m, v[c]);
    float s = 0.0f;
    #pragma unroll
    for (int c = 0; c < NCLS; ++c) s += expf(v[c] - m);
    float ls = logf(s);
    #pragma unroll
    for (int c = 0; c < NCLS; ++c) out[i * NCLS + c] = v[c] - m - ls;
}

extern "C" void kernel_launch(void* const* d_in, const int* in_sizes, int n_in,
                              void* d_out, int out_size, void* d_ws, size_t ws_size,
                              hipStream_t stream) {
    const float* x  = (const float*)d_in[0];
    const int*   ei = (const int*)d_in[1];
    const float* W1 = (const float*)d_in[2];
    const float* b1 = (const float*)d_in[3];
    const float* W2 = (const float*)d_in[4];
    const float* b2 = (const float*)d_in[5];
    float* out = (float*)d_out;

    const int N = in_sizes[0] / FIN;
    const long long E = in_sizes[1] / 2;
    const int* srcIdx = ei;        // edge_index[0]
    const int* dstIdx = ei + E;    // edge_index[1]

    // workspace layout (floats): deg/dinv[N] | hlin[16N] | acc1/h1[16N] | h2lin[16N] | acc2[8N]
    float* ws    = (float*)d_ws;
    float* deg   = ws;
    float* hlin  = ws + N;
    float* acc1  = hlin + 16LL * N;
    float* h2lin = acc1 + 16LL * N;
    float* acc2  = h2lin + 16LL * N;

    dim3 blk(TPB);
    unsigned gInit  = (unsigned)((16LL * N + TPB - 1) / TPB);
    unsigned gEdge  = (unsigned)((E + TPB - 1) / TPB);
    unsigned gNode  = (unsigned)((N + TPB - 1) / TPB);
    long long tiles = (N + 15) / 16;
    unsigned gGemm  = (unsigned)((tiles + (TPB / 32) - 1) / (TPB / 32));

    gcn_init_kernel<<<gInit, blk, 0, stream>>>(deg, acc1, acc2, N);
    gcn_degree_kernel<<<gEdge, blk, 0, stream>>>(dstIdx, deg, E);
    gcn_rsqrt_kernel<<<gNode, blk, 0, stream>>>(deg, N);
    gcn_gemm1_kernel<<<gGemm, blk, 0, stream>>>(x, W1, hlin, N);
    gcn_scatter1_kernel<<<gEdge, blk, 0, stream>>>(srcIdx, dstIdx, deg, hlin, acc1, E);
    gcn_post1_kernel<<<gInit, blk, 0, stream>>>(deg, hlin, b1, acc1, N);
    gcn_gemm2_kernel<<<gGemm, blk, 0, stream>>>(acc1, W2, h2lin, N);
    gcn_scatter2_kernel<<<gEdge, blk, 0, stream>>>(srcIdx, dstIdx, deg, h2lin, acc2, E);
    gcn_final_kernel<<<gNode, blk, 0, stream>>>(deg, h2lin, acc2, b2, out, N);
}